// GCNLayer_17257178596103
// MI455X (gfx1250) — compile-verified
//
#include <hip/hip_runtime.h>
#include <hip/hip_bf16.h>

#define N_NODES 100000
#define DIM 128
#define LN_EPS 1e-5f

typedef __attribute__((ext_vector_type(2))) float v2f;
typedef __attribute__((ext_vector_type(8))) float v8f;

#define ROWS_PER_BLOCK 64
#define LDS_STRIDE 132   // 132 % 64 == 4 -> conflict-free WMMA A-fragment reads

// ---------------- Kernel 0: zero the accumulator (d_out) ----------------
__global__ void gcn_zero_kernel(float4* __restrict__ out, int n4) {
    int i = blockIdx.x * blockDim.x + threadIdx.x;
    if (i < n4) out[i] = make_float4(0.f, 0.f, 0.f, 0.f);
}

// ------- Kernel 1: fused LayerNorm + Linear, H = LN(X) @ W^T + b --------
// Block: 256 threads (8 wave32), 64 nodes. Each wave owns a 16-wide output
// column strip (o0 = wave*16) and iterates 4 row tiles, reusing B fragments.
__global__ __launch_bounds__(256) void gcn_ln_gemm_kernel(
    const float* __restrict__ X, const float* __restrict__ gamma,
    const float* __restrict__ beta, const float* __restrict__ W,
    const float* __restrict__ bias, float* __restrict__ H) {
    __shared__ float lds[ROWS_PER_BLOCK * LDS_STRIDE];
    const int t = threadIdx.x;
    const int block_row0 = blockIdx.x * ROWS_PER_BLOCK;
    const bool full_block = (block_row0 + ROWS_PER_BLOCK) <= N_NODES;  // uniform

    // Phase 1: coalesced global->LDS copy of the 64x128 X tile (row-padded).
    if (full_block) {
        #pragma unroll
        for (int i = 0; i < 32; ++i) {
            int idx = i * 256 + t;
            int r = idx >> 7, c = idx & 127;
            lds[r * LDS_STRIDE + c] = X[(size_t)(block_row0 + r) * DIM + c];
        }
    } else {
        #pragma unroll
        for (int i = 0; i < 32; ++i) {
            int idx = i * 256 + t;
            int r = idx >> 7, c = idx & 127;
            int gr = block_row0 + r;
            int sr = gr < N_NODES ? gr : (N_NODES - 1);
            lds[r * LDS_STRIDE + c] = X[(size_t)sr * DIM + c];
        }
    }
    __syncthreads();

    // Phase 2: LayerNorm in LDS. Thread t: row t/4, cols (t%4)*32 .. +31.
    {
        int r  = t >> 2;
        int c0 = (t & 3) * 32;
        float* row = &lds[r * LDS_STRIDE];
        float s = 0.f, s2 = 0.f;
        #pragma unroll
        for (int c = 0; c < 32; ++c) { float v = row[c0 + c]; s += v; s2 += v * v; }
        // reduce over the 4-lane group within the wave32
        s  += __shfl_xor(s,  1, 32);  s2 += __shfl_xor(s2, 1, 32);
        s  += __shfl_xor(s,  2, 32);  s2 += __shfl_xor(s2, 2, 32);
        float mu   = s * (1.0f / DIM);
        float var  = s2 * (1.0f / DIM) - mu * mu;
        float rstd = rsqrtf(var + LN_EPS);
        #pragma unroll
        for (int c = 0; c < 32; ++c) {
            int cc = c0 + c;
            float v = row[cc];
            row[cc] = (v - mu) * rstd * gamma[cc] + beta[cc];
        }
    }
    __syncthreads();

    // Phase 3: GEMM via V_WMMA_F32_16X16X4_F32.
    const int wave = t >> 5;
    const int lane = t & 31;
    const int half = lane >> 4;      // 0: K pair {0,1}, 1: K pair {2,3}
    const int l15  = lane & 15;      // A: row m, B: col n
    const int o0   = wave * 16;      // output column base

    v8f acc[4];
    {
        float bv = bias[o0 + l15];   // C[m][n] = b[n] folds bias into accumulate
        #pragma unroll
        for (int tt = 0; tt < 4; ++tt)
            #pragma unroll
            for (int v = 0; v < 8; ++v) acc[tt][v] = bv;
    }

    const float* wrow = &W[(size_t)(o0 + l15) * DIM];  // B[k][n] = W[o0+n][k]
    #pragma unroll 4
    for (int k0 = 0; k0 < DIM; k0 += 4) {
        int k = k0 + 2 * half;
        v2f b; b.x = wrow[k]; b.y = wrow[k + 1];
        #pragma unroll
        for (int tt = 0; tt < 4; ++tt) {
            const float* arow = &lds[(tt * 16 + l15) * LDS_STRIDE];
            v2f a; a.x = arow[k]; a.y = arow[k + 1];
            acc[tt] = __builtin_amdgcn_wmma_f32_16x16x4_f32(
                false, a, false, b, (short)0, acc[tt], false, false);
        }
    }

    // Store: VGPR v -> row (v + 8*half), cols o0 + l15 (two 64B runs / store).
    if (full_block) {
        #pragma unroll
        for (int tt = 0; tt < 4; ++tt) {
            size_t m_base = (size_t)(block_row0 + tt * 16 + half * 8);
            float* hp = H + m_base * DIM + o0 + l15;
            #pragma unroll
            for (int v = 0; v < 8; ++v) hp[(size_t)v * DIM] = acc[tt][v];
        }
    } else {
        #pragma unroll
        for (int tt = 0; tt < 4; ++tt) {
            int m_base = block_row0 + tt * 16 + half * 8;
            #pragma unroll
            for (int v = 0; v < 8; ++v) {
                int gr = m_base + v;
                if (gr < N_NODES) H[(size_t)gr * DIM + o0 + l15] = acc[tt][v];
            }
        }
    }
}

// -------- Kernel 2: edge-parallel SpMM scatter, one wave32 per edge ------
// H rows (51MB) are L2-resident on MI455X (192MB L2): gathers + f32 atomics
// are served at L2, never HBM.
__global__ __launch_bounds__(256) void gcn_spmm_kernel(
    const float* __restrict__ H, const int* __restrict__ edge_src,
    const int* __restrict__ edge_dst, const float* __restrict__ edge_w,
    float* __restrict__ out, int n_edges) {
    int gtid = blockIdx.x * blockDim.x + threadIdx.x;
    int e    = gtid >> 5;
    int lane = threadIdx.x & 31;
    if (e >= n_edges) return;
    int   src = edge_src[e];   // uniform across wave -> scalar broadcast
    int   dst = edge_dst[e];
    float w   = edge_w[e];
    const float4* hs = (const float4*)(H + (size_t)src * DIM);
    float4 hv = hs[lane];                 // coalesced B128, 512B per wave
    float* o = out + (size_t)dst * DIM + lane * 4;
    atomicAdd(o + 0, w * hv.x);           // global_atomic_add_f32, no return
    atomicAdd(o + 1, w * hv.y);
    atomicAdd(o + 2, w * hv.z);
    atomicAdd(o + 3, w * hv.w);
}

// ---------------- Kernel 3: out = relu(out) + X (in place) ---------------
__global__ void gcn_relu_res_kernel(float4* __restrict__ out,
                                    const float4* __restrict__ X, int n4) {
    int i = blockIdx.x * blockDim.x + threadIdx.x;
    if (i < n4) {
        float4 a = out[i];
        float4 x = X[i];
        float4 r;
        r.x = (a.x > 0.f ? a.x : 0.f) + x.x;
        r.y = (a.y > 0.f ? a.y : 0.f) + x.y;
        r.z = (a.z > 0.f ? a.z : 0.f) + x.z;
        r.w = (a.w > 0.f ? a.w : 0.f) + x.w;
        out[i] = r;
    }
}

extern "C" void kernel_launch(void* const* d_in, const int* in_sizes, int n_in,
                              void* d_out, int out_size, void* d_ws, size_t ws_size,
                              hipStream_t stream) {
    const float* X        = (const float*)d_in[0];
    const int*   edge_src = (const int*)  d_in[1];
    const int*   edge_dst = (const int*)  d_in[2];
    const float* edge_w   = (const float*)d_in[3];
    const float* gamma    = (const float*)d_in[4];
    const float* beta     = (const float*)d_in[5];
    const float* W        = (const float*)d_in[6];
    const float* bias     = (const float*)d_in[7];
    float* out = (float*)d_out;
    float* H   = (float*)d_ws;                 // 100000*128*4 = 51.2 MB scratch

    const int n_edges = in_sizes[3];           // edge_w element count
    const int n_elem  = N_NODES * DIM;
    const int n4      = n_elem / 4;            // 12.8M divisible by 4

    gcn_zero_kernel<<<(n4 + 255) / 256, 256, 0, stream>>>((float4*)out, n4);

    gcn_ln_gemm_kernel<<<(N_NODES + ROWS_PER_BLOCK - 1) / ROWS_PER_BLOCK,
                         256, 0, stream>>>(X, gamma, beta, W, bias, H);

    long long spmm_threads = (long long)n_edges * 32;
    gcn_spmm_kernel<<<(int)((spmm_threads + 255) / 256), 256, 0, stream>>>(
        H, edge_src, edge_dst, edge_w, out, n_edges);

    gcn_relu_res_kernel<<<(n4 + 255) / 256, 256, 0, stream>>>(
        (float4*)out, (const float4*)X, n4);
}